// SoilModelCQ_76373108457546
// MI455X (gfx1250) — compile-verified
//
#include <hip/hip_runtime.h>
#include <cstdint>
#include <cstddef>

// Problem sizes (fixed by the reference).
#define NT_ 2000
#define NS_ 512
#define NH_ 64

// Forcing staging: chunks of 50 timesteps (2000 = 40 * 50), double buffered.
#define CHUNK 50
#define NCHUNK (NT_ / CHUNK)
#define RPITCH (CHUNK + 1)   // 51 float4 = 816B row pitch: 816%256=48,
                             // gcd(48,256)=16 -> 16 writer lanes, 16 banks.

#if __has_builtin(__builtin_amdgcn_permlane16) && __has_builtin(__builtin_amdgcn_permlanex16)
#define HAS_PERMLANE 1
#else
#define HAS_PERMLANE 0
#endif

// ---------------- CDNA5 helpers ----------------

// Low 32 bits of a flat pointer to __shared__ memory == LDS byte offset.
__device__ __forceinline__ uint32_t lds_byte_off(const void* p) {
  return (uint32_t)(uintptr_t)p;
}

// Async global -> LDS copy, 4 bytes per active lane. Tracked by ASYNCcnt.
// GV addressing mode (ISA 15.18.3 opcode 96, GLOBAL_LOAD_ASYNC_TO_LDS_B32).
__device__ __forceinline__ void async_ld_f32(uint32_t lds_off, const float* g) {
  asm volatile("global_load_async_to_lds_b32 %0, %1, off"
               :: "v"(lds_off), "v"(g)
               : "memory");
}

__device__ __forceinline__ void wait_async_le3() {
#if __has_builtin(__builtin_amdgcn_s_wait_asynccnt)
  __builtin_amdgcn_s_wait_asynccnt(3);
#else
  asm volatile("s_wait_asynccnt 3" ::: "memory");
#endif
}

__device__ __forceinline__ void wait_async_le0() {
#if __has_builtin(__builtin_amdgcn_s_wait_asynccnt)
  __builtin_amdgcn_s_wait_asynccnt(0);
#else
  asm volatile("s_wait_asynccnt 0" ::: "memory");
#endif
}

#if HAS_PERMLANE
// v_permlanex16_b32, identity selector: swap the two 16-lane halves (xor 16).
__device__ __forceinline__ float plx16(float v) {
  int i = __float_as_int(v);
  return __int_as_float(
      __builtin_amdgcn_permlanex16(i, i, 0x76543210u, 0xFEDCBA98u, false, false));
}
// v_permlane16_b32 gather within 16-lane rows; selector nibble i = i^m.
template <uint32_t S0, uint32_t S1>
__device__ __forceinline__ float pl16(float v) {
  int i = __float_as_int(v);
  return __int_as_float(__builtin_amdgcn_permlane16(i, i, S0, S1, false, false));
}
__device__ __forceinline__ float wave_sum1(float v) {
  v += plx16(v);
  v += pl16<0xFEDCBA98u, 0x76543210u>(v);   // xor 8
  v += pl16<0x32107654u, 0xBA98FEDCu>(v);   // xor 4
  v += pl16<0x54761032u, 0xDCFE98BAu>(v);   // xor 2
  v += pl16<0x67452301u, 0xEFCDAB89u>(v);   // xor 1
  return v;
}
#else
template <int XM>
__device__ __forceinline__ float swz_xor(float v) {
  return __int_as_float(
      __builtin_amdgcn_ds_swizzle(__float_as_int(v), (XM << 10) | 0x1f));
}
__device__ __forceinline__ float plx16(float v) { return swz_xor<16>(v); }
__device__ __forceinline__ float wave_sum1(float v) {
  v += swz_xor<16>(v); v += swz_xor<8>(v); v += swz_xor<4>(v);
  v += swz_xor<2>(v);  v += swz_xor<1>(v);
  return v;
}
#endif

// Issue async loads for forcing chunk c into buffer fb (= fbuf[b]).
// Layout: one float4 {P, T, E, pad} per timestep -> single ds_load_b128 read.
__device__ __forceinline__ void stage_chunk(int c, int tid, int site,
                                            const float* __restrict__ P,
                                            const float* __restrict__ Tm,
                                            const float* __restrict__ E,
                                            float4* fb) {
  if (tid < CHUNK) {
    size_t g = (size_t)(c * CHUNK + tid) * NS_ + (size_t)site;
    uint32_t base = lds_byte_off(&fb[tid]);
    async_ld_f32(base + 0, P + g);
    async_ld_f32(base + 4, Tm + g);
    async_ld_f32(base + 8, E + g);
  }
}

// ---------------- main kernel: one block = one site ----------------

__global__ __launch_bounds__(64)
void soilcq_fwd(const float* __restrict__ P,  const float* __restrict__ Tm,
                const float* __restrict__ E,
                const float* __restrict__ w,   const float* __restrict__ wk1,
                const float* __restrict__ wk2, const float* __restrict__ we1,
                const float* __restrict__ we2, const float* __restrict__ wl,
                const float* __restrict__ wo,  const float* __restrict__ wc,
                float* __restrict__ outQ,  float* __restrict__ outC,
                float* __restrict__ outS,  float* __restrict__ outH1,
                float* __restrict__ outH2, float* __restrict__ outQs) {
  const int site = blockIdx.x;
  const int tid  = threadIdx.x;   // tid == hidden unit index (0..63)
  const int lane = tid & 31;
  const int wv   = tid >> 5;      // wave 0 / 1 within block

  __shared__ float4 fbuf[2][CHUNK];    // [parity][t]: {P, T, E, pad}
  __shared__ float4 red[32][RPITCH];   // [partial i = wv*16+lane][step k]

  // ----- per-hidden-unit parameters -----
  const float eo = expf(wo[tid]);
  {
    float ps = wave_sum1(eo);
    if (lane == 0) red[wv * 16][0].x = ps;
  }
  __syncthreads();
  const float asum = red[0][0].x + red[16][0].x;
  __syncthreads();

  const float a   = eo / asum;                       // softmax(wo)
  const float ew  = expf(w[tid]) + 1.0f;             // snow melt factor
  const float sk1 = 1.0f / (1.0f + expf(-wk1[tid]));
  const float sk2 = 1.0f / (1.0f + expf(-wk2[tid]));
  const float se1 = 1.0f / (1.0f + expf(-we1[tid]));
  const float se2 = 1.0f / (1.0f + expf(-we2[tid]));
  const float Lc  = expf(wl[tid]);                   // layer capacity
  const float r   = 1.0f / sk2 / (1.0f + expf(wc[tid]));
  const float ar  = a * r;

  float sS = 0.0f, h1 = 0.0f, h2 = 0.0f;             // carry state

  // Prime the async pipeline (depth 2).
  stage_chunk(0, tid, site, P, Tm, E, fbuf[0]);
  stage_chunk(1, tid, site, P, Tm, E, fbuf[1]);

  const size_t step = (size_t)NS_ * NH_;             // 1 << 15 elements
  // Chunk-local bases; per-step strides become 24-bit store immediates
  // (49 * 128KB = 6.4MB < +-8MB IOFFSET range) under the unrolled loop.
  float* pS  = outS  + (size_t)site * NH_ + tid;
  float* pH1 = outH1 + (size_t)site * NH_ + tid;
  float* pH2 = outH2 + (size_t)site * NH_ + tid;

  for (int c = 0; c < NCHUNK; ++c) {
    // Wait for this chunk's 3 async loads (in-order per wave); keep the next
    // chunk's 3 in flight except at the tail. The barrier also separates the
    // previous chunk's red[] reads from this chunk's red[] writes.
    if (c + 2 < NCHUNK) wait_async_le3(); else wait_async_le0();
    __syncthreads();
    const int b = c & 1;

#pragma unroll 10
    for (int k = 0; k < CHUNK; ++k) {
      const float4 f = fbuf[b][k];     // one ds_load_b128 broadcast
      const float p  = f.x;
      const float tt = f.y;
      const float ee = f.z;

      // --- SnowBucket ---
      const float m = fminf(fmaxf(tt, 0.0f) * ew, sS);
      sS = sS - m + (tt < 0.0f ? p : 0.0f);
      const float x = (tt > 0.0f ? p : 0.0f) + m;

      // --- SoilBucket ---
      const float h2x = h2 + x;
      const float h1a = fmaxf(h1 + h2x - Lc, 0.0f);
      const float q1  = h1a * sk1;
      const float h2a = fminf(h1a + h2x, Lc);
      const float q2  = h2a * sk2;                 // q3 == q2 in reference
      h1 = fmaxf(h1a - q1 - ee * se1, 0.0f);
      h2 = fmaxf(h2a - 2.0f * q2 - ee * se2, 0.0f);

      // Streaming state outputs (write-once, no reuse -> TH_NT);
      // k*step folds into the global-store immediate offset.
      const size_t koff = (size_t)k * step;
      __builtin_nontemporal_store(sS, pS + koff);
      __builtin_nontemporal_store(h1, pH1 + koff);
      __builtin_nontemporal_store(h2, pH2 + koff);

      // --- reductions: ONE xor16 stage, park 32 partial triples in LDS.
      float r0 = q1 * a;
      float r1 = q2 * a;
      float r2 = q2 * ar;
      r0 += plx16(r0); r1 += plx16(r1); r2 += plx16(r2);
      if (lane < 16)
        red[wv * 16 + lane][k] = make_float4(r0, r1, r2, 0.0f);
    }
    pS  += (size_t)CHUNK * step;   // base pointers advance once per chunk
    pH1 += (size_t)CHUNK * step;
    pH2 += (size_t)CHUNK * step;
    __syncthreads();  // red[] complete; all fbuf[b] reads done.

    // --- chunk-amortized combine: thread k finalizes timestep c*CHUNK+k ---
    if (tid < CHUNK) {
      float Qs1 = 0.0f, Qs2 = 0.0f, s3 = 0.0f;
#pragma unroll 4
      for (int i = 0; i < 32; ++i) {
        const float4 v = red[i][tid];  // 16B lane stride: conflict-free
        Qs1 += v.x; Qs2 += v.y; s3 += v.z;
      }
      const float Qk = Qs1 + Qs2;
      const float cc = (s3 * (1.0f / NH_) + 1e-5f) / (Qk + 1e-5f);
      const size_t o = (size_t)(c * CHUNK + tid) * NS_ + (size_t)site;
      __builtin_nontemporal_store(Qk,   outQ + o);
      __builtin_nontemporal_store(cc,   outC + o);
      __builtin_nontemporal_store(Qs1,  outQs + 3 * o);
      __builtin_nontemporal_store(Qs2,  outQs + 3 * o + 1);
      __builtin_nontemporal_store(0.0f, outQs + 3 * o + 2);
    }

    // Restage fbuf[b] with chunk c+2 (reads of it finished before barrier).
    if (c + 2 < NCHUNK) stage_chunk(c + 2, tid, site, P, Tm, E, fbuf[b]);
  }
}

// ---------------- host-side launch ----------------

extern "C" void kernel_launch(void* const* d_in, const int* in_sizes, int n_in,
                              void* d_out, int out_size, void* d_ws, size_t ws_size,
                              hipStream_t stream) {
  const float* P   = (const float*)d_in[0];
  const float* Tm  = (const float*)d_in[1];
  const float* E   = (const float*)d_in[2];
  const float* w   = (const float*)d_in[3];
  const float* wk1 = (const float*)d_in[4];
  const float* wk2 = (const float*)d_in[5];
  const float* we1 = (const float*)d_in[6];
  const float* we2 = (const float*)d_in[7];
  const float* wl  = (const float*)d_in[8];
  const float* wo  = (const float*)d_in[9];
  const float* wc  = (const float*)d_in[10];

  float* out = (float*)d_out;
  const size_t nQ = (size_t)NT_ * NS_;        // (NT, NS)
  const size_t nS = (size_t)NT_ * NS_ * NH_;  // (NT, NS, NH)
  float* outQ  = out;                 // Q
  float* outC  = out + nQ;            // C
  float* outS  = out + 2 * nQ;        // S
  float* outH1 = outS + nS;           // H1
  float* outH2 = outH1 + nS;          // H2
  float* outQs = outH2 + nS;          // Qs (NT, NS, 3)

  soilcq_fwd<<<dim3(NS_), dim3(64), 0, stream>>>(
      P, Tm, E, w, wk1, wk2, we1, we2, wl, wo, wc,
      outQ, outC, outS, outH1, outH2, outQs);

  (void)in_sizes; (void)n_in; (void)out_size; (void)d_ws; (void)ws_size;
}